// DifferentiableMVO_63909113364604
// MI455X (gfx1250) — compile-verified
//
#include <hip/hip_runtime.h>
#include <hip/hip_bf16.h>

#define NASSET 512
#define POWER_ITERS 20
#define PGD_ITERS 150
#define BISECT_ITERS 50

typedef float v2f __attribute__((ext_vector_type(2)));
typedef float v8f __attribute__((ext_vector_type(8)));

__device__ __forceinline__ float wave_sum(float x) {
#pragma unroll
  for (int m = 16; m >= 1; m >>= 1) x += __shfl_xor(x, m, 32);
  return x;
}

__device__ __forceinline__ float wave_max(float x) {
#pragma unroll
  for (int m = 16; m >= 1; m >>= 1) x = fmaxf(x, __shfl_xor(x, m, 32));
  return x;
}

// out[n] = sum_k v[k] * Sigma[k][n]  (== Sigma @ v since Sigma is symmetric)
// A (16x4 f32): every row = (v[kb..kb+3])  -> two LDS broadcasts per step
// B (4x16 f32): Sigma[kb..kb+3][nb..nb+15] in the native B layout
// D row 0 (VGPR0, lanes 0-15) accumulates the 16 outputs of the n-tile.
__device__ __forceinline__ void matvec_wmma(const float* __restrict__ Sg,
                                            const float* __restrict__ vin,
                                            float* __restrict__ vout,
                                            int wave, int lane) {
  const int nb0  = wave * 32;
  const int nb1  = nb0 + 16;
  const int rsel = (lane >> 4) << 1;  // lanes 0-15 -> rows kb,kb+1 ; lanes 16-31 -> kb+2,kb+3
  const int col  = lane & 15;
  v8f c0 = {};
  v8f c1 = {};
#pragma unroll 4
  for (int kb = 0; kb < NASSET; kb += 4) {
    const int rk = kb + rsel;
    v2f a;
    a.x = vin[rk];
    a.y = vin[rk + 1];
    const float* p = Sg + (size_t)rk * NASSET;
    if ((kb & 15) == 0) {
      // speculative prefetch 16 rows ahead (global_prefetch_b8; OOB is dropped)
      __builtin_prefetch(p + 16 * NASSET + nb0 + col, 0, 0);
    }
    v2f b0, b1;
    b0.x = p[nb0 + col];
    b0.y = p[NASSET + nb0 + col];
    b1.x = p[nb1 + col];
    b1.y = p[NASSET + nb1 + col];
    c0 = __builtin_amdgcn_wmma_f32_16x16x4_f32(false, a, false, b0, (short)0, c0, false, false);
    c1 = __builtin_amdgcn_wmma_f32_16x16x4_f32(false, a, false, b1, (short)0, c1, false, false);
  }
  if (lane < 16) {
    vout[nb0 + lane] = c0[0];
    vout[nb1 + lane] = c1[0];
  }
}

// Projection onto {y >= 0, mu.y = 1}; executed by wave 0 only.
// Each lane holds 16 components in registers; 50 bisection steps with
// butterfly-shuffle reductions, zero workgroup barriers inside the loop.
__device__ __forceinline__ void project_wave(float (&vv)[16], const float (&mu)[16],
                                             float* __restrict__ ldsV, int lane) {
  float mx = -3.402823e38f, dmv = 0.f, dmm = 0.f;
#pragma unroll
  for (int j = 0; j < 16; ++j) {
    mx  = fmaxf(mx, vv[j] / mu[j]);
    dmv = fmaf(mu[j], vv[j], dmv);
    dmm = fmaf(mu[j], mu[j], dmm);
  }
  mx  = wave_max(mx);
  dmv = wave_sum(dmv);
  dmm = wave_sum(dmm);
  float lo = -mx;                                   // h(lo) = 0
  float hi = fmaxf(lo + 1.0f, (1.0f - dmv) / dmm);  // h(hi) >= 1
#pragma unroll 1
  for (int it = 0; it < BISECT_ITERS; ++it) {
    const float mid = 0.5f * (lo + hi);
    float h = 0.f;
#pragma unroll
    for (int j = 0; j < 16; ++j)
      h = fmaf(mu[j], fmaxf(fmaf(mid, mu[j], vv[j]), 0.f), h);
    h = wave_sum(h);  // wave-uniform
    if (h < 1.0f) lo = mid; else hi = mid;
  }
  const float lam = 0.5f * (lo + hi);
#pragma unroll
  for (int j = 0; j < 16; ++j)
    ldsV[j * 32 + lane] = fmaxf(fmaf(lam, mu[j], vv[j]), 0.f);
}

__global__ void __launch_bounds__(512)
mvo_solve_kernel(const float* __restrict__ mu_g,
                 const float* __restrict__ Sg_all,
                 float* __restrict__ out_g) {
  __shared__ float ldsMu[NASSET];
  __shared__ float ldsV[NASSET];
  __shared__ float ldsOut[NASSET];

  const int b    = blockIdx.x;
  const int tid  = threadIdx.x;
  const int wave = tid >> 5;
  const int lane = tid & 31;
  const float* Sg = Sg_all + (size_t)b * NASSET * NASSET;

  ldsMu[tid] = mu_g[(size_t)b * NASSET + tid];
  ldsV[tid]  = 1.0f;
  __syncthreads();

  const bool w0 = (wave == 0);
  float mu_l[16];
  if (w0) {
#pragma unroll
    for (int j = 0; j < 16; ++j) mu_l[j] = ldsMu[j * 32 + lane];
  }

  // ---- power iteration: v <- Sigma v / ||Sigma v|| ----
  for (int it = 0; it < POWER_ITERS; ++it) {
    matvec_wmma(Sg, ldsV, ldsOut, wave, lane);
    __syncthreads();
    if (w0) {
      float o[16];
      float ss = 0.f;
#pragma unroll
      for (int j = 0; j < 16; ++j) {
        o[j] = ldsOut[j * 32 + lane];
        ss   = fmaf(o[j], o[j], ss);
      }
      ss = wave_sum(ss);
      const float nrm = sqrtf(ss);
#pragma unroll
      for (int j = 0; j < 16; ++j) ldsV[j * 32 + lane] = o[j] / nrm;
    }
    __syncthreads();
  }

  // ---- lmax = v . (Sigma v); step = 1/lmax; initial y = project(ones) ----
  matvec_wmma(Sg, ldsV, ldsOut, wave, lane);
  __syncthreads();
  float step = 0.f;
  if (w0) {
    float d = 0.f;
#pragma unroll
    for (int j = 0; j < 16; ++j)
      d = fmaf(ldsV[j * 32 + lane], ldsOut[j * 32 + lane], d);
    d = wave_sum(d);
    step = 1.0f / d;
    float vv[16];
#pragma unroll
    for (int j = 0; j < 16; ++j) vv[j] = 1.0f;
    project_wave(vv, mu_l, ldsV, lane);
  }
  __syncthreads();

  // ---- projected gradient descent ----
  for (int it = 0; it < PGD_ITERS; ++it) {
    matvec_wmma(Sg, ldsV, ldsOut, wave, lane);
    __syncthreads();
    if (w0) {
      float vv[16];
#pragma unroll
      for (int j = 0; j < 16; ++j)
        vv[j] = fmaf(-step, ldsOut[j * 32 + lane], ldsV[j * 32 + lane]);
      project_wave(vv, mu_l, ldsV, lane);
    }
    __syncthreads();
  }

  // ---- post-processing: valid fallback, normalize, relu, renormalize ----
  if (w0) {
    bool anyp = false;
#pragma unroll
    for (int j = 0; j < 16; ++j) anyp = anyp || (mu_l[j] > 1e-6f);
    const bool valid = (__ballot(anyp) != 0ull);
    float y[16];
    float s = 0.f;
#pragma unroll
    for (int j = 0; j < 16; ++j) {
      y[j] = valid ? ldsV[j * 32 + lane] : 1.0f;
      s += y[j];
    }
    s = wave_sum(s);
    const bool oks = fabsf(s) > 1e-6f;
    float w[16];
    float s2 = 0.f;
#pragma unroll
    for (int j = 0; j < 16; ++j) {
      w[j] = oks ? (y[j] / s) : (1.0f / NASSET);
      w[j] = fmaxf(w[j], 0.f);
      s2 += w[j];
    }
    s2 = wave_sum(s2);
    float* og = out_g + (size_t)b * NASSET;
#pragma unroll
    for (int j = 0; j < 16; ++j) og[j * 32 + lane] = w[j] / s2;
  }
}

extern "C" void kernel_launch(void* const* d_in, const int* in_sizes, int n_in,
                              void* d_out, int out_size, void* d_ws, size_t ws_size,
                              hipStream_t stream) {
  (void)n_in; (void)out_size; (void)d_ws; (void)ws_size;
  const float* mu = (const float*)d_in[0];    // [B, 512] f32
  const float* Sg = (const float*)d_in[1];    // [B, 512, 512] f32 (symmetric)
  float* out = (float*)d_out;                 // [B, 512] f32
  const int batch = in_sizes[0] / NASSET;
  mvo_solve_kernel<<<dim3(batch), dim3(512), 0, stream>>>(mu, Sg, out);
}